// multimodal_RGCN_40724879900960
// MI455X (gfx1250) — compile-verified
//
#include <hip/hip_runtime.h>
#include <hip/hip_bf16.h>

typedef __attribute__((ext_vector_type(2))) float v2f;
typedef __attribute__((ext_vector_type(8))) float v8f;

#define DIM 128
#define SLOPE 0.01f

__device__ __forceinline__ float lrelu(float v) { return v >= 0.f ? v : SLOPE * v; }
__device__ __forceinline__ void atomAddF(float* p, float v) { unsafeAtomicAdd(p, v); }

// ---------------------------------------------------------------------------
// C[M x 128] = A[M x 128] @ B[128 x 128], fp32 WMMA (16x16x4). Grid = M/128
// blocks of 256 threads (8 waves, one 16-row strip per wave).
// B staged in LDS pair-interleaved: Bs[(row>>1)*256 + col*2 + (row&1)] so each
// WMMA B-fragment {B[k+2hg][n], B[k+2hg+1][n]} is one aligned ds_load_b64.
// A strip prefetched entirely into registers (32 x v2f) to avoid per-step
// global-load stalls in the WMMA loop.
// ---------------------------------------------------------------------------
__global__ __launch_bounds__(256) void gemm128_wmma(const float* __restrict__ A,
                                                    const float* __restrict__ B,
                                                    float* __restrict__ C) {
  __shared__ float Bs[DIM * DIM];
  const int t = threadIdx.x;
#pragma unroll
  for (int i = 0; i < 64; ++i) {
    int idx = t + i * 256;
    int row = idx >> 7, col = idx & 127;
    Bs[(row >> 1) * 256 + col * 2 + (row & 1)] = B[idx];
  }
  __syncthreads();

  const int lane = t & 31;
  const int wave = t >> 5;
  const int lm = lane & 15;   // row/col within 16
  const int hg = lane >> 4;   // half-wave group
  const size_t m0 = (size_t)blockIdx.x * 128 + wave * 16;
  const float* Arow = A + (m0 + lm) * DIM + 2 * hg;

  // prefetch entire per-lane A strip: a[kk] = A[m0+lm][4kk+2hg .. 4kk+2hg+1]
  v2f a[32];
#pragma unroll
  for (int kk = 0; kk < 32; ++kk) a[kk] = *(const v2f*)(Arow + kk * 4);

  const v8f vz = {0.f, 0.f, 0.f, 0.f, 0.f, 0.f, 0.f, 0.f};
  v8f acc[8];
#pragma unroll
  for (int ti = 0; ti < 8; ++ti) acc[ti] = vz;

#pragma unroll
  for (int kk = 0; kk < 32; ++kk) {
    const float* bbase = &Bs[(kk * 2 + hg) * 256 + lm * 2];
#pragma unroll
    for (int ti = 0; ti < 8; ++ti) {
      v2f b = *(const v2f*)(bbase + ti * 32);
      acc[ti] = __builtin_amdgcn_wmma_f32_16x16x4_f32(
          false, a[kk], false, b, (short)0, acc[ti], false, false);
    }
  }
#pragma unroll
  for (int ti = 0; ti < 8; ++ti)
#pragma unroll
    for (int r = 0; r < 8; ++r)
      C[(m0 + r + 8 * hg) * DIM + ti * 16 + lm] = acc[ti][r];
}

// ---------------------------------------------------------------------------
// Hn[r] = X[r] * rsqrt(sum(X[r]^2) + 1e-8) ; one wave per row
// ---------------------------------------------------------------------------
__global__ __launch_bounds__(256) void row_normalize(const float* __restrict__ X,
                                                     float* __restrict__ Hn, int Nrows) {
  int r = blockIdx.x * 8 + (threadIdx.x >> 5);
  if (r >= Nrows) return;
  int lane = threadIdx.x & 31;
  float4 v = ((const float4*)(X + (size_t)r * DIM))[lane];
  float ss = v.x * v.x + v.y * v.y + v.z * v.z + v.w * v.w;
  for (int o = 16; o; o >>= 1) ss += __shfl_xor(ss, o, 32);
  float s = rsqrtf(ss + 1e-8f);
  float4 ov = make_float4(v.x * s, v.y * s, v.z * s, v.w * s);
  ((float4*)(Hn + (size_t)r * DIM))[lane] = ov;
}

// out[i] = bias[i % 128]  (accumulator init with GCN bias)
__global__ void init_bias(float* __restrict__ out, const float* __restrict__ b, size_t n) {
  for (size_t i = (size_t)blockIdx.x * blockDim.x + threadIdx.x; i < n;
       i += (size_t)gridDim.x * blockDim.x)
    out[i] = b[i & (DIM - 1)];
}

// out[dst] += ew[e] * M[src]; one wave per edge, float4 per lane
__global__ __launch_bounds__(256) void edge_scatter(const float* __restrict__ M,
                                                    const int* __restrict__ src,
                                                    const int* __restrict__ dst,
                                                    const float* __restrict__ ew,
                                                    float* __restrict__ out, int E) {
  int e = blockIdx.x * 8 + (threadIdx.x >> 5);
  if (e >= E) return;
  int lane = threadIdx.x & 31;
  size_t s = (size_t)src[e] * DIM, d = (size_t)dst[e] * DIM;
  float w = ew[e];
  float4 v = ((const float4*)(M + s))[lane];
  float* od = out + d + lane * 4;
  atomAddF(od + 0, v.x * w);
  atomAddF(od + 1, v.y * w);
  atomAddF(od + 2, v.z * w);
  atomAddF(od + 3, v.w * w);
}

// ew[e] = 0.5*dot(Hn[s],Hn[d]) + 0.5*sigmoid(dot(G[s],H0[d]))   (A only read at edges)
__global__ __launch_bounds__(256) void edge_weight(const float* __restrict__ Hn,
                                                   const float* __restrict__ G,
                                                   const float* __restrict__ H0,
                                                   const int* __restrict__ src,
                                                   const int* __restrict__ dst,
                                                   float* __restrict__ ew, int E) {
  int e = blockIdx.x * 8 + (threadIdx.x >> 5);
  if (e >= E) return;
  int lane = threadIdx.x & 31;
  size_t s = (size_t)src[e] * DIM, d = (size_t)dst[e] * DIM;
  float4 a = ((const float4*)(Hn + s))[lane];
  float4 b = ((const float4*)(Hn + d))[lane];
  float4 g = ((const float4*)(G + s))[lane];
  float4 h = ((const float4*)(H0 + d))[lane];
  float d0 = a.x * b.x + a.y * b.y + a.z * b.z + a.w * b.w;
  float d1 = g.x * h.x + g.y * h.y + g.z * h.z + g.w * h.w;
  for (int o = 16; o; o >>= 1) {
    d0 += __shfl_xor(d0, o, 32);
    d1 += __shfl_xor(d1, o, 32);
  }
  if (lane == 0) ew[e] = 0.5f * d0 + 0.5f * (1.f / (1.f + __expf(-d1)));
}

// H = lrelu(a + b + c), vectorized
__global__ void combine_lrelu(const float4* __restrict__ a, const float4* __restrict__ b,
                              const float4* __restrict__ c, float4* __restrict__ o, size_t n4) {
  for (size_t i = (size_t)blockIdx.x * blockDim.x + threadIdx.x; i < n4;
       i += (size_t)gridDim.x * blockDim.x) {
    float4 va = a[i], vb = b[i], vc = c[i];
    float4 r;
    r.x = lrelu(va.x + vb.x + vc.x);
    r.y = lrelu(va.y + vb.y + vc.y);
    r.z = lrelu(va.z + vb.z + vc.z);
    r.w = lrelu(va.w + vb.w + vc.w);
    o[i] = r;
  }
}

// per-column mean & inv-std (biased var) over R rows; grid = C blocks of 128
__global__ __launch_bounds__(128) void col_stats(const float* __restrict__ X,
                                                 float* __restrict__ mean,
                                                 float* __restrict__ inv, int R, int C) {
  int c = blockIdx.x;
  float s = 0.f, s2 = 0.f;
  for (int r = threadIdx.x; r < R; r += 128) {
    float v = X[(size_t)r * C + c];
    s += v;
    s2 += v * v;
  }
  __shared__ float sh[128], sh2[128];
  sh[threadIdx.x] = s;
  sh2[threadIdx.x] = s2;
  __syncthreads();
  for (int o = 64; o; o >>= 1) {
    if (threadIdx.x < o) {
      sh[threadIdx.x] += sh[threadIdx.x + o];
      sh2[threadIdx.x] += sh2[threadIdx.x + o];
    }
    __syncthreads();
  }
  if (threadIdx.x == 0) {
    float m = sh[0] / R;
    float var = sh2[0] / R - m * m;
    mean[c] = m;
    inv[c] = rsqrtf(var + 1e-5f);
  }
}

// x = g*(x-m)*inv + b   (in place)
__global__ void bn_apply(float* __restrict__ x, const float* __restrict__ m,
                         const float* __restrict__ inv, const float* __restrict__ g,
                         const float* __restrict__ b, size_t n, int C) {
  for (size_t i = (size_t)blockIdx.x * blockDim.x + threadIdx.x; i < n;
       i += (size_t)gridDim.x * blockDim.x) {
    int c = (int)(i % C);
    x[i] = g[c] * (x[i] - m[c]) * inv[c] + b[c];
  }
}

// gather p,q; comb=[p,q,p-q,p*q]; z=lrelu(comb@W1+b1); h5=z@W2+b2. grid=B*5, block=128
__global__ __launch_bounds__(128) void tail_gf(const float* __restrict__ Hbn,
                                               const int* __restrict__ post_idx,
                                               const int* __restrict__ image_idx,
                                               const float* __restrict__ W1,
                                               const float* __restrict__ b1,
                                               const float* __restrict__ W2,
                                               const float* __restrict__ b2,
                                               float* __restrict__ h5) {
  int bk = blockIdx.x, b = bk / 5, k = bk % 5, t = threadIdx.x;
  __shared__ float p[DIM], q[DIM], z[DIM];
  int pi = post_idx[b * 5 + k], qi = image_idx[b * 5 + k];
  p[t] = Hbn[(size_t)(b * 32 + pi) * DIM + t];
  q[t] = Hbn[(size_t)(b * 32 + qi) * DIM + t];
  __syncthreads();
  float acc = b1[t];
  for (int c = 0; c < DIM; ++c) {
    float pc = p[c], qc = q[c];
    acc += pc * W1[c * DIM + t] + qc * W1[(DIM + c) * DIM + t] +
           (pc - qc) * W1[(2 * DIM + c) * DIM + t] + (pc * qc) * W1[(3 * DIM + c) * DIM + t];
  }
  z[t] = lrelu(acc);
  __syncthreads();
  if (t < 64) {
    float a2 = b2[t];
    for (int c = 0; c < DIM; ++c) a2 += z[c] * W2[c * 64 + t];
    h5[(size_t)bk * 64 + t] = a2;
  }
}

// BN(h5) -> lrelu(@am_W1) -> @am_W2 ; grid=B, block=128
__global__ __launch_bounds__(128) void tail_am(const float* __restrict__ h5,
                                               const float* __restrict__ m,
                                               const float* __restrict__ inv,
                                               const float* __restrict__ g,
                                               const float* __restrict__ bb,
                                               const float* __restrict__ W1,
                                               const float* __restrict__ b1,
                                               const float* __restrict__ W2,
                                               const float* __restrict__ b2,
                                               float* __restrict__ h2) {
  int r = blockIdx.x, t = threadIdx.x;
  __shared__ float hn[320], z[128];
  for (int c = t; c < 320; c += 128) {
    float v = h5[(size_t)r * 320 + c];
    hn[c] = g[c] * (v - m[c]) * inv[c] + bb[c];
  }
  __syncthreads();
  float acc = b1[t];
  for (int c = 0; c < 320; ++c) acc += hn[c] * W1[c * 128 + t];
  z[t] = lrelu(acc);
  __syncthreads();
  if (t < 64) {
    float a2 = b2[t];
    for (int c = 0; c < 128; ++c) a2 += z[c] * W2[c * 64 + t];
    h2[(size_t)r * 64 + t] = a2;
  }
}

// BN(h2) -> lrelu(@fl_W1) -> @fl_W2 = out ; prob = out@out_W + out_b ; grid=B, block=64
__global__ __launch_bounds__(64) void tail_fl(const float* __restrict__ h2,
                                              const float* __restrict__ m,
                                              const float* __restrict__ inv,
                                              const float* __restrict__ g,
                                              const float* __restrict__ bb,
                                              const float* __restrict__ W1,
                                              const float* __restrict__ b1,
                                              const float* __restrict__ W2,
                                              const float* __restrict__ b2,
                                              const float* __restrict__ oW,
                                              const float* __restrict__ ob,
                                              float* __restrict__ out, float* __restrict__ prob) {
  int r = blockIdx.x, t = threadIdx.x;
  __shared__ float hn[64], u[64], o[32];
  {
    float v = h2[(size_t)r * 64 + t];
    hn[t] = g[t] * (v - m[t]) * inv[t] + bb[t];
  }
  __syncthreads();
  float acc = b1[t];
  for (int c = 0; c < 64; ++c) acc += hn[c] * W1[c * 64 + t];
  u[t] = lrelu(acc);
  __syncthreads();
  if (t < 32) {
    float a2 = b2[t];
    for (int c = 0; c < 64; ++c) a2 += u[c] * W2[c * 32 + t];
    o[t] = a2;
    out[(size_t)r * 32 + t] = a2;
  }
  __syncthreads();
  if (t < 2) {
    float a3 = ob[t];
    for (int c = 0; c < 32; ++c) a3 += o[c] * oW[c * 2 + t];
    prob[r * 2 + t] = a3;
  }
}

extern "C" void kernel_launch(void* const* d_in, const int* in_sizes, int n_in,
                              void* d_out, int out_size, void* d_ws, size_t ws_size,
                              hipStream_t stream) {
  const float* x = (const float*)d_in[0];
  const int* eiA = (const int*)d_in[1];
  const float* eaA = (const float*)d_in[2];
  const int* eiP = (const int*)d_in[3];
  const float* eaP = (const float*)d_in[4];
  const int* eiI = (const int*)d_in[5];
  const float* eaI = (const float*)d_in[6];
  const int* post_idx = (const int*)d_in[7];
  const int* image_idx = (const int*)d_in[8];
  const float* Wg = (const float*)d_in[10];   // [2,3,128,128]
  const float* bg = (const float*)d_in[11];   // [2,3,128]
  const float* dynW = (const float*)d_in[12]; // [2,128,128]
  const float* hbn_g = (const float*)d_in[13];
  const float* hbn_b = (const float*)d_in[14];
  const float* gfW1 = (const float*)d_in[15];
  const float* gfb1 = (const float*)d_in[16];
  const float* gfW2 = (const float*)d_in[17];
  const float* gfb2 = (const float*)d_in[18];
  const float* bng = (const float*)d_in[19];
  const float* bnb = (const float*)d_in[20];
  const float* amW1 = (const float*)d_in[21];
  const float* amb1 = (const float*)d_in[22];
  const float* amW2 = (const float*)d_in[23];
  const float* amb2 = (const float*)d_in[24];
  const float* bn2g = (const float*)d_in[25];
  const float* bn2b = (const float*)d_in[26];
  const float* flW1 = (const float*)d_in[27];
  const float* flb1 = (const float*)d_in[28];
  const float* flW2 = (const float*)d_in[29];
  const float* flb2 = (const float*)d_in[30];
  const float* outW = (const float*)d_in[31];
  const float* outb = (const float*)d_in[32];

  const int N = in_sizes[0] / DIM;        // 4096
  const int B = in_sizes[7] / 5;          // 128
  const int EA = in_sizes[2], EP = in_sizes[4], EI = in_sizes[6];
  const size_t ND = (size_t)N * DIM;

  float* w = (float*)d_ws;
  float* Hn = w;    w += ND;
  float* M = w;     w += ND;
  float* Hall = w;  w += ND;
  float* Hpost = w; w += ND;
  float* Himg = w;  w += ND;
  float* Hcur = w;  w += ND;
  float* G = w;     w += ND;
  float* ewA = w;   w += EA;
  float* ewP = w;   w += EP;
  float* ewI = w;   w += EI;
  float* mean128 = w; w += 128;
  float* inv128 = w;  w += 128;
  float* mean320 = w; w += 320;
  float* inv320 = w;  w += 320;
  float* mean64 = w;  w += 64;
  float* inv64 = w;   w += 64;
  float* h5 = w;    w += (size_t)B * 320;
  float* h2 = w;    w += (size_t)B * 64;

  float* out = (float*)d_out;
  float* prob = out + (size_t)B * 32;

  const int GB = 2048, GT = 256;        // grid-stride config
  const int gemmBlocks = N / 128;

  // cosine-similarity operand (from original x)
  row_normalize<<<(N + 7) / 8, 256, 0, stream>>>(x, Hn, N);

  // ---- layer 0 (edge weights = ea inputs) ----
  gemm128_wmma<<<gemmBlocks, 256, 0, stream>>>(x, Wg + 0 * 16384, M);
  init_bias<<<GB, GT, 0, stream>>>(Hall, bg + 0, ND);
  edge_scatter<<<(EA + 7) / 8, 256, 0, stream>>>(M, eiA, eiA + EA, eaA, Hall, EA);
  gemm128_wmma<<<gemmBlocks, 256, 0, stream>>>(Hall, Wg + 1 * 16384, M);
  init_bias<<<GB, GT, 0, stream>>>(Hpost, bg + 128, ND);
  edge_scatter<<<(EP + 7) / 8, 256, 0, stream>>>(M, eiP, eiP + EP, eaP, Hpost, EP);
  gemm128_wmma<<<gemmBlocks, 256, 0, stream>>>(Hall, Wg + 2 * 16384, M);
  init_bias<<<GB, GT, 0, stream>>>(Himg, bg + 256, ND);
  edge_scatter<<<(EI + 7) / 8, 256, 0, stream>>>(M, eiI, eiI + EI, eaI, Himg, EI);
  combine_lrelu<<<GB, GT, 0, stream>>>((const float4*)Hall, (const float4*)Hpost,
                                       (const float4*)Himg, (float4*)Hcur, ND / 4);

  // ---- layer-1 edge weights: A evaluated only at edge (src,dst) pairs ----
  gemm128_wmma<<<gemmBlocks, 256, 0, stream>>>(Hcur, dynW, G);
  edge_weight<<<(EA + 7) / 8, 256, 0, stream>>>(Hn, G, Hcur, eiA, eiA + EA, ewA, EA);
  edge_weight<<<(EP + 7) / 8, 256, 0, stream>>>(Hn, G, Hcur, eiP, eiP + EP, ewP, EP);
  edge_weight<<<(EI + 7) / 8, 256, 0, stream>>>(Hn, G, Hcur, eiI, eiI + EI, ewI, EI);

  // ---- layer 1 ----
  gemm128_wmma<<<gemmBlocks, 256, 0, stream>>>(Hcur, Wg + 3 * 16384, M);
  init_bias<<<GB, GT, 0, stream>>>(Hall, bg + 384, ND);
  edge_scatter<<<(EA + 7) / 8, 256, 0, stream>>>(M, eiA, eiA + EA, ewA, Hall, EA);
  gemm128_wmma<<<gemmBlocks, 256, 0, stream>>>(Hall, Wg + 4 * 16384, M);
  init_bias<<<GB, GT, 0, stream>>>(Hpost, bg + 512, ND);
  edge_scatter<<<(EP + 7) / 8, 256, 0, stream>>>(M, eiP, eiP + EP, ewP, Hpost, EP);
  gemm128_wmma<<<gemmBlocks, 256, 0, stream>>>(Hall, Wg + 5 * 16384, M);
  init_bias<<<GB, GT, 0, stream>>>(Himg, bg + 640, ND);
  edge_scatter<<<(EI + 7) / 8, 256, 0, stream>>>(M, eiI, eiI + EI, ewI, Himg, EI);
  combine_lrelu<<<GB, GT, 0, stream>>>((const float4*)Hall, (const float4*)Hpost,
                                       (const float4*)Himg, (float4*)Hcur, ND / 4);

  // ---- BatchNorm over nodes ----
  col_stats<<<DIM, 128, 0, stream>>>(Hcur, mean128, inv128, N, DIM);
  bn_apply<<<GB, GT, 0, stream>>>(Hcur, mean128, inv128, hbn_g, hbn_b, ND, DIM);

  // ---- per-graph head ----
  tail_gf<<<B * 5, 128, 0, stream>>>(Hcur, post_idx, image_idx, gfW1, gfb1, gfW2, gfb2, h5);
  col_stats<<<320, 128, 0, stream>>>(h5, mean320, inv320, B, 320);
  tail_am<<<B, 128, 0, stream>>>(h5, mean320, inv320, bng, bnb, amW1, amb1, amW2, amb2, h2);
  col_stats<<<64, 128, 0, stream>>>(h2, mean64, inv64, B, 64);
  tail_fl<<<B, 64, 0, stream>>>(h2, mean64, inv64, bn2g, bn2b, flW1, flb1, flW2, flb2,
                                outW, outb, out, prob);
}